// MeshFeatureExtractor_76871324664398
// MI455X (gfx1250) — compile-verified
//
#include <hip/hip_runtime.h>

#define NN 100000
#define EE 3200000
#define EPSF 1e-5f

typedef float v2f __attribute__((ext_vector_type(2)));
typedef float v8f __attribute__((ext_vector_type(8)));

// ---------------- utility ----------------
__global__ void fill_f32(float* __restrict__ p, float v, int n) {
  int i = blockIdx.x * blockDim.x + threadIdx.x;
  if (i < n) p[i] = v;
}

// deg[col[e]] += w[e]   (deg pre-filled with 1.0 == self-loop weight)
__global__ void deg_kernel(const int* __restrict__ col, const float* __restrict__ w,
                           float* __restrict__ deg, int ne) {
  int e = blockIdx.x * blockDim.x + threadIdx.x;
  if (e < ne) atomicAdd(&deg[col[e]], w[e]);
}

// deg -> dinv in place; selfn = dinv^2  (self-loop norm, weight 1)
__global__ void dinv_kernel(float* __restrict__ deg, float* __restrict__ selfn, int n) {
  int i = blockIdx.x * blockDim.x + threadIdx.x;
  if (i < n) {
    float di = rsqrtf(deg[i]);   // deg >= 1 always (self loop)
    deg[i] = di;
    selfn[i] = di * di;
  }
}

__global__ void norm_kernel(const int* __restrict__ row, const int* __restrict__ col,
                            const float* __restrict__ w, const float* __restrict__ dinv,
                            float* __restrict__ norm, int ne) {
  int e = blockIdx.x * blockDim.x + threadIdx.x;
  if (e < ne) norm[e] = dinv[row[e]] * w[e] * dinv[col[e]];
}

// ---------------- batchnorm ----------------
// stats[0..C-1] = sum, stats[C..2C-1] = sumsq (stats pre-zeroed). blockDim must be 256.
__global__ void bn_stats(const float* __restrict__ h, int stride, int C, int CH, int n,
                         float* __restrict__ stats) {
  int c = threadIdx.x % CH;
  int rgrp = threadIdx.x / CH;
  int nrg = 256 / CH;
  float s = 0.f, q = 0.f;
  if (c < C) {
    for (int i = blockIdx.x * nrg + rgrp; i < n; i += gridDim.x * nrg) {
      float v = h[(size_t)i * stride + c];
      s += v; q += v * v;
    }
  }
  __shared__ float sh[512];
  sh[threadIdx.x] = s;
  sh[256 + threadIdx.x] = q;
  __syncthreads();
  if (threadIdx.x < (unsigned)CH && c < C) {
    float ts = 0.f, tq = 0.f;
    for (int r = 0; r < nrg; ++r) { ts += sh[r * CH + c]; tq += sh[256 + r * CH + c]; }
    atomicAdd(&stats[c], ts);
    atomicAdd(&stats[C + c], tq);
  }
}

// BN of raw input x (13 ch) -> h0 padded to stride 16 (cols 13..15 = 0)
__global__ void bn_apply_x(const float* __restrict__ x, const float* __restrict__ stats,
                           const float* __restrict__ g, const float* __restrict__ be,
                           float* __restrict__ h0, int n) {
  int idx = blockIdx.x * blockDim.x + threadIdx.x;
  int row = idx >> 4, c = idx & 15;
  if (row >= n) return;
  float out = 0.f;
  if (c < 13) {
    float inv_n = 1.f / (float)n;
    float mu = stats[c] * inv_n;
    float var = stats[13 + c] * inv_n - mu * mu;
    out = (x[row * 13 + c] - mu) * rsqrtf(var + EPSF) * g[c] + be[c];
  }
  h0[row * 16 + c] = out;
}

// in-place BN for 32-channel buffers
__global__ void bn_apply32(float* __restrict__ h, const float* __restrict__ stats,
                           const float* __restrict__ g, const float* __restrict__ be, int n) {
  int idx = blockIdx.x * blockDim.x + threadIdx.x;
  if (idx >= n * 32) return;
  int c = idx & 31;
  float inv_n = 1.f / (float)n;
  float mu = stats[c] * inv_n;
  float var = stats[32 + c] * inv_n - mu * mu;
  h[idx] = (h[idx] - mu) * rsqrtf(var + EPSF) * g[c] + be[c];
}

__global__ void bias_relu(float* __restrict__ h, const float* __restrict__ b, int n) {
  int idx = blockIdx.x * blockDim.x + threadIdx.x;
  if (idx >= n * 32) return;
  h[idx] = fmaxf(h[idx] + b[idx & 31], 0.f);
}

// zero-pad W1 (13x32) to 16x32
__global__ void pad_w1(const float* __restrict__ W1, float* __restrict__ W1p) {
  int t = threadIdx.x;            // 512 threads
  int k = t >> 5;
  W1p[t] = (k < 13) ? W1[t] : 0.f;
}

// ---------------- WMMA fp32 GEMM: C[n x 32] = A[n x Kd] * B[Kd x 32] ----------------
// One wave per 16-row tile; V_WMMA_F32_16X16X4_F32, two 16-col halves.
// Requires n % 16 == 0 (N = 100000 = 6250*16). EXEC all-1s per wave.
__global__ void __launch_bounds__(256)
gcn_gemm_wmma(const float* __restrict__ A, int lda, int Kd,
              const float* __restrict__ B, float* __restrict__ C, int nrows) {
  const int lane = threadIdx.x & 31;
  const int wv = threadIdx.x >> 5;
  const int tile = blockIdx.x * 8 + wv;
  const int row0 = tile * 16;
  if (row0 >= nrows) return;            // wave-uniform exit
  const int m = lane & 15;              // row (A) / col (B,C) within tile
  const int g = lane >> 4;              // lane group
  const float* arow = A + (size_t)(row0 + m) * lda + 2 * g;
  v8f acc0 = {0.f, 0.f, 0.f, 0.f, 0.f, 0.f, 0.f, 0.f};
  v8f acc1 = acc0;
  for (int ks = 0; ks < Kd; ks += 4) {
    // A 16x4: lanegroup g, VGPR r -> K = ks + 2g + r
    v2f a;
    a.x = arow[ks];
    a.y = arow[ks + 1];
    // B 4x16: lanegroup g, VGPR r -> K = ks + 2g + r, N = m (+16 for half 1)
    const float* bp = B + (size_t)(ks + 2 * g) * 32 + m;
    v2f b0; b0.x = bp[0];  b0.y = bp[32];
    v2f b1; b1.x = bp[16]; b1.y = bp[48];
    acc0 = __builtin_amdgcn_wmma_f32_16x16x4_f32(false, a, false, b0, (short)0, acc0, false, false);
    acc1 = __builtin_amdgcn_wmma_f32_16x16x4_f32(false, a, false, b1, (short)0, acc1, false, false);
  }
  // D 16x16 f32: VGPR r -> M = r + 8g, N = m
  float* crow = C + (size_t)(row0 + 8 * g) * 32 + m;
#pragma unroll
  for (int r = 0; r < 8; ++r) {
    crow[(size_t)r * 32 + 0]  = acc0[r];
    crow[(size_t)r * 32 + 16] = acc1[r];
  }
}

// ---------------- message passing ----------------
// agg[i] = selfn[i] * hW[i]   (writes, so no pre-zero of agg needed)
__global__ void self_msg(const float* __restrict__ hW, const float* __restrict__ selfn,
                         float* __restrict__ agg, int n) {
  int idx = blockIdx.x * blockDim.x + threadIdx.x;   // n*8 threads
  int i = idx >> 3, c4 = (idx & 7) * 4;
  if (i >= n) return;
  float s = selfn[i];
  const float4 v = *(const float4*)(hW + (size_t)i * 32 + c4);
  float4 o = make_float4(s * v.x, s * v.y, s * v.z, s * v.w);
  *(float4*)(agg + (size_t)i * 32 + c4) = o;
}

// agg[col[e]] += norm[e] * hW[row[e]]  (L2-resident: hW/agg are 12.8MB each)
__global__ void edge_msg(const float* __restrict__ hW, const int* __restrict__ row,
                         const int* __restrict__ col, const float* __restrict__ norm,
                         float* __restrict__ agg, int ne) {
  int idx = blockIdx.x * blockDim.x + threadIdx.x;   // ne*8 threads
  int e = idx >> 3, c4 = (idx & 7) * 4;
  if (e >= ne) return;
  float nm = norm[e];
  const float4 v = *(const float4*)(hW + (size_t)row[e] * 32 + c4);
  float* o = agg + (size_t)col[e] * 32 + c4;
  atomicAdd(o + 0, nm * v.x);
  atomicAdd(o + 1, nm * v.y);
  atomicAdd(o + 2, nm * v.z);
  atomicAdd(o + 3, nm * v.w);
}

// ---------------- pooling ----------------
__global__ void pool_acc(const float* __restrict__ h, const int* __restrict__ batch,
                         float* __restrict__ pooled, float* __restrict__ cnt, int n) {
  int idx = blockIdx.x * blockDim.x + threadIdx.x;   // n*8 threads
  int i = idx >> 3, c4 = (idx & 7) * 4;
  if (i >= n) return;
  int gph = batch[i];
  const float4 v = *(const float4*)(h + (size_t)i * 32 + c4);
  float* p = pooled + gph * 32 + c4;
  atomicAdd(p + 0, v.x);
  atomicAdd(p + 1, v.y);
  atomicAdd(p + 2, v.z);
  atomicAdd(p + 3, v.w);
  if (c4 == 0) atomicAdd(&cnt[gph], 1.f);
}

__global__ void pool_div(const float* __restrict__ pooled, const float* __restrict__ cnt,
                         float* __restrict__ out) {
  int t = threadIdx.x;  // 256
  out[t] = pooled[t] / fmaxf(cnt[t >> 5], 1.f);
}

// ---------------- host ----------------
extern "C" void kernel_launch(void* const* d_in, const int* in_sizes, int n_in,
                              void* d_out, int out_size, void* d_ws, size_t ws_size,
                              hipStream_t stream) {
  const int n = NN, ne = EE;
  const float* x     = (const float*)d_in[0];
  const int*   erow  = (const int*)d_in[1];
  const int*   ecol  = erow + ne;
  const float* ew    = (const float*)d_in[2];
  const int*   batch = (const int*)d_in[3];
  const float* W[4]  = {(const float*)d_in[4], (const float*)d_in[6],
                        (const float*)d_in[8], (const float*)d_in[10]};
  const float* bb[4] = {(const float*)d_in[5], (const float*)d_in[7],
                        (const float*)d_in[9], (const float*)d_in[11]};
  const float* gam[5]; const float* bet[5];
  for (int i = 0; i < 5; ++i) {
    gam[i] = (const float*)d_in[12 + 2 * i];
    bet[i] = (const float*)d_in[13 + 2 * i];
  }

  float* ws    = (float*)d_ws;
  float* deg   = ws;                        // N (becomes dinv in-place)
  float* selfn = deg + n;                   // N
  float* norme = selfn + n;                 // E
  float* hb0   = norme + ne;                // N*32
  float* hb1   = hb0 + (size_t)n * 32;      // N*32
  float* hb2   = hb1 + (size_t)n * 32;      // N*32
  float* stats = hb2 + (size_t)n * 32;      // 64
  float* W1p   = stats + 64;                // 512
  float* pooled= W1p + 512;                 // 256
  float* cnt   = pooled + 256;              // 8
  float* bufs[3] = {hb0, hb1, hb2};

  const int T = 256;
  auto blk = [](long long cnt_, int t) { return (int)((cnt_ + t - 1) / t); };

  // --- gcn_norm precompute ---
  fill_f32<<<blk(n, T), T, 0, stream>>>(deg, 1.0f, n);                  // self-loop weight
  deg_kernel<<<blk(ne, T), T, 0, stream>>>(ecol, ew, deg, ne);
  dinv_kernel<<<blk(n, T), T, 0, stream>>>(deg, selfn, n);
  norm_kernel<<<blk(ne, T), T, 0, stream>>>(erow, ecol, ew, deg, norme, ne);

  // --- BN0 on x -> padded h0 (stride 16) ---
  fill_f32<<<1, 64, 0, stream>>>(stats, 0.f, 64);
  bn_stats<<<512, T, 0, stream>>>(x, 13, 13, 16, n, stats);
  bn_apply_x<<<blk((long long)n * 16, T), T, 0, stream>>>(x, stats, gam[0], bet[0], hb0, n);
  pad_w1<<<1, 512, 0, stream>>>(W[0], W1p);

  // --- 4 GCN layers ---
  int ia = 0, lda = 16, Kd = 16;
  const float* Bmat = W1p;
  const int gemm_blocks = (n / 16 + 7) / 8;
  for (int l = 0; l < 4; ++l) {
    float* hin = bufs[ia];
    float* hW  = bufs[(ia + 1) % 3];
    float* agg = bufs[(ia + 2) % 3];
    gcn_gemm_wmma<<<gemm_blocks, T, 0, stream>>>(hin, lda, Kd, Bmat, hW, n);
    self_msg<<<blk((long long)n * 8, T), T, 0, stream>>>(hW, selfn, agg, n);
    edge_msg<<<blk((long long)ne * 8, T), T, 0, stream>>>(hW, erow, ecol, norme, agg, ne);
    bias_relu<<<blk((long long)n * 32, T), T, 0, stream>>>(agg, bb[l], n);
    fill_f32<<<1, 64, 0, stream>>>(stats, 0.f, 64);
    bn_stats<<<512, T, 0, stream>>>(agg, 32, 32, 32, n, stats);
    bn_apply32<<<blk((long long)n * 32, T), T, 0, stream>>>(agg, stats, gam[l + 1], bet[l + 1], n);
    ia = (ia + 2) % 3;
    lda = 32; Kd = 32;
    if (l + 1 < 4) Bmat = W[l + 1];
  }

  // --- global mean pool ---
  fill_f32<<<2, 160, 0, stream>>>(pooled, 0.f, 264);   // pooled(256) + cnt(8)
  pool_acc<<<blk((long long)n * 8, T), T, 0, stream>>>(bufs[ia], batch, pooled, cnt, n);
  pool_div<<<1, 256, 0, stream>>>(pooled, cnt, (float*)d_out);
}